// GCN_53455162966134
// MI455X (gfx1250) — compile-verified
//
#include <hip/hip_runtime.h>
#include <hip/hip_bf16.h>
#include <math.h>

typedef __attribute__((ext_vector_type(2))) float v2f;
typedef __attribute__((ext_vector_type(8))) float v8f;

#define F_IN 128
#define HID 16
#define N_CLS 40

// ---------------- degree / normalization ----------------
__global__ void deg_init_k(float* __restrict__ deg, int n) {
    int i = blockIdx.x * 256 + threadIdx.x;
    if (i < n) deg[i] = 1.0f;                 // self-loop contributes 1
}

__global__ void deg_count_k(const int* __restrict__ col, float* __restrict__ deg, int E) {
    int e = blockIdx.x * 256 + threadIdx.x;
    if (e < E) atomicAdd(&deg[col[e]], 1.0f);
}

__global__ void deg_rsqrt_k(float* __restrict__ deg, int n) {
    int i = blockIdx.x * 256 + threadIdx.x;
    if (i < n) deg[i] = rsqrtf(deg[i]);       // deg >= 1 always (self loops)
}

// ---------------- GEMM1: h1[N,16] = x[N,128] @ W1[128,16] via v_wmma_f32_16x16x4_f32 ----------
__global__ void gemm1_wmma_k(const float* __restrict__ x, const float* __restrict__ W1,
                             float* __restrict__ h1, int n, int n_tiles) {
    int wave = (blockIdx.x * blockDim.x + threadIdx.x) >> 5;   // one 16-row tile per wave
    int lane = threadIdx.x & 31;
    if (wave >= n_tiles) return;                                // wave-uniform exit
    int row0 = wave * 16;
    int half = lane >> 4;          // 0: lanes 0-15, 1: lanes 16-31
    int mn   = lane & 15;          // M index for A, N index for B/C/D

    int arow = row0 + mn; if (arow >= n) arow = n - 1;          // clamp loads for ragged tail
    const float* xrow = x + (size_t)arow * F_IN + 2 * half;     // K offset per half-wave

    v8f c = {0.f,0.f,0.f,0.f,0.f,0.f,0.f,0.f};
    #pragma unroll 4
    for (int k = 0; k < F_IN; k += 4) {
        // A 16x4 f32: VGPR0 = K=k (lo lanes) / K=k+2 (hi lanes), VGPR1 = +1
        v2f a = *(const v2f*)(xrow + k);
        // B 4x16 f32: N on lanes, K rows split across halves like A
        v2f b;
        b.x = W1[(k + 2 * half) * HID + mn];
        b.y = W1[(k + 2 * half + 1) * HID + mn];
        c = __builtin_amdgcn_wmma_f32_16x16x4_f32(false, a, false, b, (short)0, c, false, false);
    }
    // C/D: VGPR r -> M = r + 8*half, N = mn
    float* hp = h1 + (size_t)(row0 + 8 * half) * HID + mn;
    if (row0 + 16 <= n) {                       // wave-uniform fast path: 8 plain stores
        #pragma unroll
        for (int r = 0; r < 8; ++r) hp[(size_t)r * HID] = c[r];
    } else {                                    // ragged tail (unused for N=100000)
        #pragma unroll
        for (int r = 0; r < 8; ++r)
            if (row0 + r + 8 * half < n) hp[(size_t)r * HID] = c[r];
    }
}

// ---------------- self-loop + bias init, layer 1 ----------------
__global__ void init_agg1_k(const float* __restrict__ h1, const float* __restrict__ dinv,
                            const float* __restrict__ b1, float* __restrict__ agg1, int n) {
    int t = blockIdx.x * 256 + threadIdx.x;
    if (t < n * HID) {
        int i = t >> 4, f = t & 15;
        float di = dinv[i];
        agg1[t] = b1[f] + h1[t] * di * di;
    }
}

// ---------------- edge scatter, layer 1 (16 lanes per edge, coalesced gather) ------------
__global__ void scatter1_k(const int* __restrict__ row, const int* __restrict__ col,
                           const float* __restrict__ h1, const float* __restrict__ dinv,
                           float* __restrict__ agg1, int E) {
    int t = blockIdx.x * 256 + threadIdx.x;
    if (t >= E * HID) return;
    int e = t >> 4, f = t & 15;
    int r = row[e], c = col[e];
    float v = h1[(size_t)r * HID + f] * dinv[r] * dinv[c];
    atomicAdd(&agg1[(size_t)c * HID + f], v);
}

// ---------------- GEMM2: h2[N,40] = relu(agg1)[N,16] @ W2[16,40], N padded to 48 ---------
__global__ void gemm2_wmma_k(const float* __restrict__ agg1, const float* __restrict__ W2,
                             float* __restrict__ h2, int n, int n_mtiles) {
    int mtile = blockIdx.x * (blockDim.x >> 5) + (threadIdx.x >> 5);
    int ntile = blockIdx.y;                     // 0..2 (N padded 40 -> 48)
    int lane  = threadIdx.x & 31;
    if (mtile >= n_mtiles) return;              // wave-uniform exit
    int row0 = mtile * 16;
    int half = lane >> 4;
    int mn   = lane & 15;
    int ncol = ntile * 16 + mn;
    bool nok = (ncol < N_CLS);

    int arow = row0 + mn; if (arow >= n) arow = n - 1;
    const float* ar = agg1 + (size_t)arow * HID + 2 * half;

    v8f c = {0.f,0.f,0.f,0.f,0.f,0.f,0.f,0.f};
    #pragma unroll
    for (int k = 0; k < HID; k += 4) {
        v2f a = *(const v2f*)(ar + k);          // 8B-aligned: (row*16 + 2*half + k) is even
        a.x = a.x > 0.f ? a.x : 0.f;            // fused ReLU on layer-1 activations
        a.y = a.y > 0.f ? a.y : 0.f;
        int kk = k + 2 * half;
        v2f b;
        b.x = nok ? W2[kk * N_CLS + ncol] : 0.f;
        b.y = nok ? W2[(kk + 1) * N_CLS + ncol] : 0.f;
        c = __builtin_amdgcn_wmma_f32_16x16x4_f32(false, a, false, b, (short)0, c, false, false);
    }
    float* hp = h2 + (size_t)(row0 + 8 * half) * N_CLS + ncol;
    if (row0 + 16 <= n) {                       // wave-uniform; single exec mask for ncol<40
        if (nok) {
            #pragma unroll
            for (int r = 0; r < 8; ++r) hp[(size_t)r * N_CLS] = c[r];
        }
    } else {
        #pragma unroll
        for (int r = 0; r < 8; ++r)
            if (nok && (row0 + r + 8 * half < n)) hp[(size_t)r * N_CLS] = c[r];
    }
}

// ---------------- self-loop + bias init, layer 2 (writes into d_out) ----------------
__global__ void init_agg2_k(const float* __restrict__ h2, const float* __restrict__ dinv,
                            const float* __restrict__ b2, float* __restrict__ agg2, int n) {
    int t = blockIdx.x * 256 + threadIdx.x;
    if (t < n * N_CLS) {
        int i = t / N_CLS, f = t - i * N_CLS;
        float di = dinv[i];
        agg2[t] = b2[f] + h2[t] * di * di;
    }
}

// ---------------- edge scatter, layer 2 (40 feats per edge) ----------------
__global__ void scatter2_k(const int* __restrict__ row, const int* __restrict__ col,
                           const float* __restrict__ h2, const float* __restrict__ dinv,
                           float* __restrict__ agg2, int E) {
    int t = blockIdx.x * 256 + threadIdx.x;
    if (t >= E * N_CLS) return;
    int e = t / N_CLS, f = t - e * N_CLS;
    int r = row[e], c = col[e];
    float v = h2[(size_t)r * N_CLS + f] * dinv[r] * dinv[c];
    atomicAdd(&agg2[(size_t)c * N_CLS + f], v);
}

// ---------------- log_softmax over 40 classes, one wave32 per node, in place ------------
__global__ void logsoftmax_k(float* __restrict__ out, int n) {
    int node = blockIdx.x * (blockDim.x >> 5) + (threadIdx.x >> 5);
    int lane = threadIdx.x & 31;
    if (node >= n) return;
    float* p = out + (size_t)node * N_CLS;
    float v0 = p[lane];                                   // lanes 0..31 -> classes 0..31
    bool has2 = (lane + 32) < N_CLS;                      // lanes 0..7 -> classes 32..39
    float v1 = has2 ? p[lane + 32] : -INFINITY;
    float m = fmaxf(v0, v1);
    #pragma unroll
    for (int off = 16; off > 0; off >>= 1) m = fmaxf(m, __shfl_xor(m, off, 32));
    float s = expf(v0 - m) + (has2 ? expf(v1 - m) : 0.f);
    #pragma unroll
    for (int off = 16; off > 0; off >>= 1) s += __shfl_xor(s, off, 32);
    float lse = m + logf(s);
    p[lane] = v0 - lse;
    if (has2) p[lane + 32] = v1 - lse;
}

extern "C" void kernel_launch(void* const* d_in, const int* in_sizes, int n_in,
                              void* d_out, int out_size, void* d_ws, size_t ws_size,
                              hipStream_t stream) {
    const float* x  = (const float*)d_in[0];
    const int*   ei = (const int*)d_in[1];
    const float* W1 = (const float*)d_in[2];
    const float* b1 = (const float*)d_in[3];
    const float* W2 = (const float*)d_in[4];
    const float* b2 = (const float*)d_in[5];

    const int N = in_sizes[0] / F_IN;       // 100000
    const int E = in_sizes[1] / 2;          // 3200000
    const int* row = ei;                    // edge_index[0]
    const int* col = ei + E;                // edge_index[1]

    // workspace layout (floats): dinv | h1[N*16] | agg1[N*16] | h2[N*40]  (~29.2 MB)
    float* wsf  = (float*)d_ws;
    float* dinv = wsf;
    float* h1   = wsf + (((size_t)N + 127) & ~(size_t)127);
    float* agg1 = h1 + (size_t)N * HID;
    float* h2   = agg1 + (size_t)N * HID;
    float* out  = (float*)d_out;            // doubles as agg2

    const int T = 256;
    // 1) symmetric normalization
    deg_init_k  <<<(N + T - 1) / T, T, 0, stream>>>(dinv, N);
    deg_count_k <<<(E + T - 1) / T, T, 0, stream>>>(col, dinv, E);
    deg_rsqrt_k <<<(N + T - 1) / T, T, 0, stream>>>(dinv, N);

    // 2) layer 1 linear (WMMA f32)
    int mtiles = (N + 15) / 16;                              // 6250
    gemm1_wmma_k<<<(mtiles + 7) / 8, T, 0, stream>>>(x, W1, h1, N, mtiles);

    // 3) layer 1 aggregate
    init_agg1_k<<<((size_t)N * HID + T - 1) / T, T, 0, stream>>>(h1, dinv, b1, agg1, N);
    scatter1_k <<<((size_t)E * HID + T - 1) / T, T, 0, stream>>>(row, col, h1, dinv, agg1, E);

    // 4) layer 2 linear on relu(agg1) (WMMA f32, 3 n-tiles for 40->48 cols)
    dim3 g2((mtiles + 7) / 8, 3);
    gemm2_wmma_k<<<g2, T, 0, stream>>>(agg1, W2, h2, N, mtiles);

    // 5) layer 2 aggregate into d_out
    init_agg2_k<<<((size_t)N * N_CLS + T - 1) / T, T, 0, stream>>>(h2, dinv, b2, out, N);
    scatter2_k <<<((size_t)E * N_CLS + T - 1) / T, T, 0, stream>>>(row, col, h2, dinv, out, E);

    // 6) log_softmax in place (one wave32 per node)
    logsoftmax_k<<<(N + 7) / 8, T, 0, stream>>>(out, N);
}